// ADJ_SecondLayer_11493332484389
// MI455X (gfx1250) — compile-verified
//
#include <hip/hip_runtime.h>
#include <hip/hip_bf16.h>

typedef __attribute__((ext_vector_type(2))) float v2f;
typedef __attribute__((ext_vector_type(4))) float v4f;
typedef __attribute__((ext_vector_type(8))) float v8f;

#define Bq    8192
#define Cq    2048
#define Lq    6
#define NRGB  4
#define HALFB 4096
#define Nq    (Bq + Lq)     // 8198
#define PAIRS (Nq / 2)      // 4099
#define NCHUNK 32
#define ROWS_PER_CHUNK (Bq / NCHUNK)  // 256

// workspace layout (float element offsets)
#define WS_SCORES 0
#define WS_W      8192
#define WS_DSAMP  16384
#define WS_DENOM  24576
#define WS_CAMD   24582
#define WS_COUNTS 24592   // int[6] stored here
#define WS_PART   24704   // NCHUNK*Lq*Cq floats

// ---------------------------------------------------------------------------
// Kernel A: scores[m] = x[m,:] . att_w[cams[m],:] + att_b[cams[m]]
// Full [B,16] = x @ W_pad^T via V_WMMA_F32_16X16X4_F32 (fp32-exact matrix
// path), one wave per 16-sample tile. Software-pipelined: next K-chunk's A/B
// loads are issued before the current WMMA so each iteration has one chunk
// of load latency in flight on top of 8-waves/block TLP.
// ---------------------------------------------------------------------------
__global__ void score_wmma_kernel(const float* __restrict__ x,
                                  const int*   __restrict__ cams,
                                  const float* __restrict__ attw,
                                  const float* __restrict__ attb,
                                  float*       __restrict__ scores) {
    const int lane    = threadIdx.x & 31;
    const int wave    = blockIdx.x * (blockDim.x >> 5) + (threadIdx.x >> 5);
    const int m0      = wave * 16;          // 512 tiles cover B=8192
    const int halfSel = lane >> 4;          // 0: lanes 0-15, 1: lanes 16-31
    const int nl      = lane & 15;
    const int koff    = halfSel * 2;        // A/B layout: second lane-half holds K+2,K+3

    const int   mA    = m0 + nl;            // row of x this lane loads for A
    const float* xr   = x + (size_t)mA * Cq + koff;
    const float bmask = (nl < Lq) ? 1.0f : 0.0f;                        // zero-pad cams 6..15
    const float* wr   = attw + (size_t)(nl < Lq ? nl : 0) * Cq + koff;  // B[k][n] = W_pad[n][k]

    v2f a, b;
    a.x = xr[0];          a.y = xr[1];
    b.x = bmask * wr[0];  b.y = bmask * wr[1];

    v8f c = {};
    for (int k = 0; k < Cq; k += 4) {
        v2f an = {}, bn = {};
        const int kn = k + 4;
        if (kn < Cq) {                      // uniform branch (k is scalar)
            an.x = xr[kn];          an.y = xr[kn + 1];
            bn.x = bmask * wr[kn];  bn.y = bmask * wr[kn + 1];
        }
        c = __builtin_amdgcn_wmma_f32_16x16x4_f32(
                /*neg_a=*/false, a, /*neg_b=*/false, b,
                /*c_mod=*/(short)0, c, /*reuse_a=*/false, /*reuse_b=*/false);
        a = an; b = bn;
    }

    // C/D layout: VGPR r holds row M=r (lanes 0-15, N=lane) and M=8+r (lanes 16-31)
    #pragma unroll
    for (int r = 0; r < 8; ++r) {
        int m  = m0 + r + 8 * halfSel;
        int cm = cams[m];
        if (nl == cm) scores[m] = c[r] + attb[cm];
    }
}

// ---------------------------------------------------------------------------
// Kernel B: deterministic per-cam denom & counts, per-sample D, cam-node D,
// and attention weights w[i] = scores[i] / denom[cams[i]].
// Single block; fixed-order two-stage reduction.
// ---------------------------------------------------------------------------
__global__ void reduce_kernel(const int* __restrict__ cams,
                              float*     __restrict__ ws) {
    __shared__ float sS[256][Lq];
    __shared__ int   sC[256][Lq];
    __shared__ float sDen[Lq];

    const int tid = threadIdx.x;
    const float* scores = ws + WS_SCORES;
    float* Dsamp = ws + WS_DSAMP;

    float ls[Lq]; int lc[Lq];
    #pragma unroll
    for (int l = 0; l < Lq; ++l) { ls[l] = 0.0f; lc[l] = 0; }

    for (int i = tid; i < Bq; i += 256) {
        int cm = cams[i];
        float s = scores[i];
        #pragma unroll
        for (int l = 0; l < Lq; ++l) {
            bool e = (cm == l);
            ls[l] += e ? s : 0.0f;
            lc[l] += e ? 1 : 0;
        }
        // sample-row degree = HALFB (bb block) + #cam nodes of other modality
        Dsamp[i] = 1.0f / sqrtf((float)HALFB + ((cm < NRGB) ? 2.0f : 4.0f));
    }
    #pragma unroll
    for (int l = 0; l < Lq; ++l) { sS[tid][l] = ls[l]; sC[tid][l] = lc[l]; }
    __syncthreads();

    if (tid == 0) {
        float dn[Lq]; int cn[Lq];
        for (int l = 0; l < Lq; ++l) { dn[l] = 0.0f; cn[l] = 0; }
        for (int t = 0; t < 256; ++t)
            for (int l = 0; l < Lq; ++l) { dn[l] += sS[t][l]; cn[l] += sC[t][l]; }
        int nrgb = 0, nir = 0;
        for (int l = 0; l < NRGB; ++l) nrgb += cn[l];
        for (int l = NRGB; l < Lq; ++l) nir += cn[l];
        float* denom = ws + WS_DENOM;
        float* camD  = ws + WS_CAMD;
        int*   cnts  = (int*)(ws + WS_COUNTS);
        for (int l = 0; l < Lq; ++l) {
            denom[l] = dn[l];
            cnts[l]  = cn[l];
            sDen[l]  = dn[l];
            // cam-node degree = 1 (cc diag) + #samples of other modality
            camD[l] = 1.0f / sqrtf(1.0f + (float)((l < NRGB) ? nir : nrgb));
        }
    }
    __syncthreads();

    float* w = ws + WS_W;
    for (int i = tid; i < Bq; i += 256) w[i] = scores[i] / sDen[cams[i]];
}

// ---------------------------------------------------------------------------
// Kernel C: partial per-cam weighted centers; chunked over rows so the final
// combine (Kernel D) is in a fixed order -> deterministic.
// ---------------------------------------------------------------------------
__global__ void center_partial_kernel(const float* __restrict__ x,
                                      const int*   __restrict__ cams,
                                      const float* __restrict__ w,
                                      float*       __restrict__ part) {
    const int c     = blockIdx.x * blockDim.x + threadIdx.x;  // column 0..2047
    const int chunk = blockIdx.y;
    float acc[Lq];
    #pragma unroll
    for (int l = 0; l < Lq; ++l) acc[l] = 0.0f;

    const int i0 = chunk * ROWS_PER_CHUNK;
    for (int r = 0; r < ROWS_PER_CHUNK; ++r) {
        int i = i0 + r;
        float wi = w[i];
        int cm = cams[i];
        float v = wi * x[(size_t)i * Cq + c];
        #pragma unroll
        for (int l = 0; l < Lq; ++l) acc[l] += (cm == l) ? v : 0.0f;
    }
    #pragma unroll
    for (int l = 0; l < Lq; ++l)
        part[(size_t)(chunk * Lq + l) * Cq + c] = acc[l];
}

// ---------------------------------------------------------------------------
// Kernel D: combine partials in fixed chunk order, select running_mean for
// absent cams, write tail rows of x_out.
// ---------------------------------------------------------------------------
__global__ void tail_kernel(const float* __restrict__ part,
                            const int*   __restrict__ cnts,
                            const float* __restrict__ rmean,
                            float*       __restrict__ xout) {
    int idx = blockIdx.x * blockDim.x + threadIdx.x;  // < Lq*Cq
    int l = idx / Cq, c = idx % Cq;
    float s = 0.0f;
    for (int ch = 0; ch < NCHUNK; ++ch)
        s += part[(size_t)(ch * Lq + l) * Cq + c];
    float out = (cnts[l] > 0) ? s : rmean[idx];
    xout[(size_t)(Bq + l) * Cq + c] = out;
}

// ---------------------------------------------------------------------------
// Kernel E: x -> x_out rows [0,B). Last consumer of x in the sequence, and
// the destination is write-once: non-temporal on both sides so the 128 MB of
// traffic doesn't sweep L2. ext_vector v4f so the NT builtins accept it.
// ---------------------------------------------------------------------------
__global__ void copy_x_kernel(const v4f* __restrict__ x4,
                              v4f*       __restrict__ out4) {
    size_t idx = (size_t)blockIdx.x * blockDim.x + threadIdx.x;
    v4f v = __builtin_nontemporal_load(&x4[idx]);
    __builtin_nontemporal_store(v, &out4[idx]);
}

// ---------------------------------------------------------------------------
// Kernel F: adjacency. adj[i][j] = D[i]*g[i][j]*D[j] with g a cheap indicator
// (same-half / cross-modality / cam diag). One block per row, coalesced
// v2f stores (row base is only 8B-aligned: 8198*4 ≡ 8 mod 16, so b64 is
// the widest legal store). 269 MB > 192 MB L2 -> non-temporal stores keep the
// stream from evicting the Dsamp/cams/x working set.
// ---------------------------------------------------------------------------
__global__ void adj_kernel(const int*   __restrict__ cams,
                           const float* __restrict__ Dsamp,
                           const float* __restrict__ camD,
                           float*       __restrict__ adj) {
    const int i = blockIdx.x;
    v2f* row = (v2f*)(adj + (size_t)i * Nq);

    if (i < Bq) {
        const bool rowFirst = i < HALFB;
        const bool rowRgb   = cams[i] < NRGB;
        const float di      = Dsamp[i];
        for (int p = threadIdx.x; p < PAIRS; p += blockDim.x) {
            int j = 2 * p;
            v2f v;
            if (j < Bq) {  // bb block: same-half indicator
                bool sh0 = ((j     < HALFB) == rowFirst);
                bool sh1 = (((j+1) < HALFB) == rowFirst);
                v.x = sh0 ? di * Dsamp[j]     : 0.0f;
                v.y = sh1 ? di * Dsamp[j + 1] : 0.0f;
            } else {       // bc block: cross-modality to cam nodes
                int n0 = j - Bq, n1 = n0 + 1;
                v.x = ((n0 < NRGB) != rowRgb) ? di * camD[n0] : 0.0f;
                v.y = ((n1 < NRGB) != rowRgb) ? di * camD[n1] : 0.0f;
            }
            __builtin_nontemporal_store(v, &row[p]);
        }
    } else {
        const int  node    = i - Bq;
        const bool nodeRgb = node < NRGB;
        const float di     = camD[node];
        for (int p = threadIdx.x; p < PAIRS; p += blockDim.x) {
            int j = 2 * p;
            v2f v;
            if (j < Bq) {  // bc^T block
                v.x = ((cams[j]     < NRGB) != nodeRgb) ? di * Dsamp[j]     : 0.0f;
                v.y = ((cams[j + 1] < NRGB) != nodeRgb) ? di * Dsamp[j + 1] : 0.0f;
            } else {       // cc block: diagonal
                v.x = ((j     - Bq) == node) ? di * di : 0.0f;
                v.y = ((j + 1 - Bq) == node) ? di * di : 0.0f;
            }
            __builtin_nontemporal_store(v, &row[p]);
        }
    }
}

extern "C" void kernel_launch(void* const* d_in, const int* in_sizes, int n_in,
                              void* d_out, int out_size, void* d_ws, size_t ws_size,
                              hipStream_t stream) {
    const float* x     = (const float*)d_in[0];
    const int*   cams  = (const int*)d_in[1];
    // d_in[2]=step, d_in[3]=modal (scalars, modal==0 branch assumed)
    const float* attw  = (const float*)d_in[4];
    const float* attb  = (const float*)d_in[5];
    const float* rmean = (const float*)d_in[6];

    float* out = (float*)d_out;
    float* ws  = (float*)d_ws;
    float* adj = out + (size_t)Nq * Cq;   // x_out first, then adj

    // A: fp32 WMMA scores GEMM (512 tiles, 8 waves/block)
    score_wmma_kernel<<<64, 256, 0, stream>>>(x, cams, attw, attb, ws + WS_SCORES);

    // B: deterministic per-cam reductions + D factors + weights
    reduce_kernel<<<1, 256, 0, stream>>>(cams, ws);

    // C: chunked partial centers
    dim3 gC(Cq / 256, NCHUNK);
    center_partial_kernel<<<gC, 256, 0, stream>>>(x, cams, ws + WS_W, ws + WS_PART);

    // D: ordered combine -> tail rows of x_out
    tail_kernel<<<(Lq * Cq) / 256, 256, 0, stream>>>(
        ws + WS_PART, (const int*)(ws + WS_COUNTS), rmean, out);

    // E: copy x into x_out (non-temporal both directions)
    copy_x_kernel<<<(Bq * Cq / 4) / 256, 256, 0, stream>>>(
        (const v4f*)x, (v4f*)out);

    // F: adjacency fill (bandwidth-dominant, NT stores)
    adj_kernel<<<Nq, 256, 0, stream>>>(cams, ws + WS_DSAMP, ws + WS_CAMD, adj);
}